// JointAttention_64957085385332
// MI455X (gfx1250) — compile-verified
//
#include <hip/hip_runtime.h>
#include <hip/hip_bf16.h>

typedef __attribute__((ext_vector_type(2))) float v2f;
typedef __attribute__((ext_vector_type(8))) float v8f;

#define DIM 256

// ---------------------------------------------------------------------------
// Kernel 1: QKV projection with V_WMMA_F32_16X16X4_F32 (full fp32 precision).
// One wave (32 threads) computes one 16x16 tile of O = X @ W^T + b.
//   grid.x = BT/16 (M tiles), grid.y = DIM/16 (N tiles), grid.z = 3 (Q/K/V)
//
// Fragment layouts (CDNA5 ISA 7.12.2, 32-bit data, wave32):
//   A (16x4):  lane L holds row m=L&15; k = 2*(L>>4) + v  (v = component 0/1)
//   B (4x16):  lane L holds col n=L&15; k = 2*(L>>4) + v
//   C/D(16x16):lane L holds col n=L&15; row = v + 8*(L>>4) (v = 0..7)
// B = W^T, i.e. B[k][n] = W[n][k], so a B fragment is a contiguous float2
// read from row n of W.
// ---------------------------------------------------------------------------
__global__ __launch_bounds__(32) void qkv_wmma_f32_kernel(
    const float* __restrict__ x,
    const float* __restrict__ Wq, const float* __restrict__ bq,
    const float* __restrict__ Wk, const float* __restrict__ bk,
    const float* __restrict__ Wv, const float* __restrict__ bv,
    float* __restrict__ Q, float* __restrict__ K, float* __restrict__ V)
{
    const float* W;
    const float* bias;
    float* O;
    if (blockIdx.z == 0)      { W = Wq; bias = bq; O = Q; }
    else if (blockIdx.z == 1) { W = Wk; bias = bk; O = K; }
    else                      { W = Wv; bias = bv; O = V; }

    const int lane  = threadIdx.x;        // 0..31 (wave32)
    const int l15   = lane & 15;
    const int khalf = lane >> 4;          // 0 or 1
    const int tm    = blockIdx.x;         // M tile
    const int tn    = blockIdx.y;         // N tile

    const float* arow = x + (size_t)(tm * 16 + l15) * DIM + 2 * khalf;
    const float* brow = W + (size_t)(tn * 16 + l15) * DIM + 2 * khalf;

    v8f c = {};
    #pragma unroll 8
    for (int k0 = 0; k0 < DIM; k0 += 4) {
        v2f a = *(const v2f*)(arow + k0);
        v2f b = *(const v2f*)(brow + k0);
        // D = A(16x4) * B(4x16) + C, fp32 everywhere
        c = __builtin_amdgcn_wmma_f32_16x16x4_f32(
                /*neg_a=*/false, a, /*neg_b=*/false, b,
                /*c_mod=*/(short)0, c, /*reuse_a=*/false, /*reuse_b=*/false);
    }

    const int   n   = tn * 16 + l15;
    const float bb  = bias[n];
    const int   mb  = tm * 16 + 8 * khalf;
    #pragma unroll
    for (int v = 0; v < 8; ++v) {
        O[(size_t)(mb + v) * DIM + n] = c[v] + bb;
    }
}

// ---------------------------------------------------------------------------
// Kernel 2: per-token scalar-outer-product softmax attention.
//   scores[i,j] = (Q_i/16) * K_j  ->  softmax over j  ->  dot with V.
// One 256-thread block per token. K,V rows staged in LDS; stable shift
// m_i = a_i * (a_i >= 0 ? max(K) : min(K)) via one block reduction.
// Inner loop: 256 x (v_exp_f32 + 2 fma) per thread.
// ---------------------------------------------------------------------------
__global__ __launch_bounds__(DIM) void outer_softmax_attn_kernel(
    const float* __restrict__ Q,
    const float* __restrict__ K,
    const float* __restrict__ V,
    float* __restrict__ out)
{
    __shared__ float sK[DIM];
    __shared__ float sV[DIM];
    __shared__ float rmax[DIM];
    __shared__ float rmin[DIM];

    const int t = blockIdx.x;
    const int i = threadIdx.x;
    const size_t base = (size_t)t * DIM;

    const float kv = K[base + i];
    sK[i]   = kv;
    sV[i]   = V[base + i];
    rmax[i] = kv;
    rmin[i] = kv;
    __syncthreads();

    #pragma unroll
    for (int s = DIM / 2; s > 0; s >>= 1) {
        if (i < s) {
            rmax[i] = fmaxf(rmax[i], rmax[i + s]);
            rmin[i] = fminf(rmin[i], rmin[i + s]);
        }
        __syncthreads();
    }
    const float kmax = rmax[0];
    const float kmin = rmin[0];

    const float a = Q[base + i] * 0.0625f;           // 1/sqrt(256)
    const float m = (a >= 0.0f) ? a * kmax : a * kmin; // max_j a*K_j

    float num = 0.0f, den = 0.0f;
    #pragma unroll 8
    for (int j = 0; j < DIM; ++j) {
        const float e = __expf(fmaf(a, sK[j], -m));
        den += e;
        num = fmaf(e, sV[j], num);
    }
    out[base + i] = num / den;
}

extern "C" void kernel_launch(void* const* d_in, const int* in_sizes, int n_in,
                              void* d_out, int out_size, void* d_ws, size_t ws_size,
                              hipStream_t stream)
{
    const float* x  = (const float*)d_in[0];
    const float* Wq = (const float*)d_in[1];
    const float* bq = (const float*)d_in[2];
    const float* Wk = (const float*)d_in[3];
    const float* bk = (const float*)d_in[4];
    const float* Wv = (const float*)d_in[5];
    const float* bv = (const float*)d_in[6];
    float* out = (float*)d_out;

    const int BT = in_sizes[0] / DIM;   // 2 * 1024 = 2048 tokens

    float* Q = (float*)d_ws;
    float* K = Q + (size_t)BT * DIM;
    float* V = K + (size_t)BT * DIM;

    // Phase 1: Q/K/V = x @ W^T + b  (WMMA fp32)
    dim3 g1(BT / 16, DIM / 16, 3);
    qkv_wmma_f32_kernel<<<g1, 32, 0, stream>>>(x, Wq, bq, Wk, bk, Wv, bv, Q, K, V);

    // Phase 2: per-token outer-product softmax + weighted V sum
    outer_softmax_attn_kernel<<<BT, DIM, 0, stream>>>(Q, K, V, out);
}